// Sampler_68719476736658
// MI455X (gfx1250) — compile-verified
//
#include <hip/hip_runtime.h>
#include <hip/hip_bf16.h>
#include <stdint.h>

#define NB      256        // batch rows
#define NV      128000     // vocab
#define TOPC    256        // top candidates kept per row (top_k <= 255)
#define HBITS   12
#define HBINS   (1 << HBITS)
#define HSHIFT  (32 - HBITS)
#define BPT     (HBINS / 256)   // histogram bins per thread
#define CAP     1024       // candidate buffer (power of 2 for bitonic)
#define TILE    1024       // floats per streamed tile (256 thr * 4)
#define NT      (NV / TILE)
#define PIPE    8          // async pipeline depth (7 tiles in flight per wave)
#define IGN     (-3000.0f)

// ---- CDNA5 async global->LDS copy (GLOBAL_LOAD_ASYNC_TO_LDS_B128, ASYNCcnt) ----
__device__ __forceinline__ void async_b128_to_lds(unsigned lds_addr, const void* g) {
    asm volatile("global_load_async_to_lds_b128 %0, %1, off"
                 :: "v"(lds_addr), "v"((unsigned long long)(uintptr_t)g)
                 : "memory");
}
__device__ __forceinline__ void wait_async_le(int rem) {
    switch (rem) {
    case 0:  asm volatile("s_wait_asynccnt 0x0" ::: "memory"); break;
    case 1:  asm volatile("s_wait_asynccnt 0x1" ::: "memory"); break;
    case 2:  asm volatile("s_wait_asynccnt 0x2" ::: "memory"); break;
    case 3:  asm volatile("s_wait_asynccnt 0x3" ::: "memory"); break;
    case 4:  asm volatile("s_wait_asynccnt 0x4" ::: "memory"); break;
    case 5:  asm volatile("s_wait_asynccnt 0x5" ::: "memory"); break;
    case 6:  asm volatile("s_wait_asynccnt 0x6" ::: "memory"); break;
    default: asm volatile("s_wait_asynccnt 0x7" ::: "memory"); break;
    }
}

__device__ __forceinline__ unsigned fkey(float f) {
    unsigned u = __float_as_uint(f);
    return u ^ ((u >> 31) ? 0xFFFFFFFFu : 0x80000000u);   // monotone: bigger float -> bigger key
}

// ================= Kernel 1: per-row top-256 (sorted) via radix-select =================
__global__ __launch_bounds__(256) void topk_kernel(const float* __restrict__ logits,
                                                   float* __restrict__ top_vals,   // [TOPC][NB] col-major
                                                   int*   __restrict__ top_idx) {  // [TOPC][NB]
    __shared__ unsigned            hist[HBINS];                 // 16 KB
    __shared__ unsigned long long  cand[CAP];                   // 8 KB
    __shared__ __align__(16) float stage[PIPE][TILE];           // 32 KB
    __shared__ unsigned            part[256];
    __shared__ int sTb;
    __shared__ int cnt;

    const int tid = threadIdx.x;
    const int row = blockIdx.x;
    const float* rp = logits + (size_t)row * NV;

    for (int i = tid; i < HBINS; i += 256) hist[i] = 0;
    if (tid == 0) cnt = 0;
    __syncthreads();

    // ---- Pass A: histogram of top 12 key bits, data streamed via async copies ----
    // Each lane consumes only the 16 B it loaded itself, so per-wave ASYNCcnt waits
    // suffice (no cross-wave barriers in the streaming loop). The stage read is
    // consumed (compiler-inserted s_wait_dscnt) one iteration before its buffer
    // is overwritten, so no explicit DS drain is needed.
    for (int t = 0; t < PIPE - 1 && t < NT; ++t)
        async_b128_to_lds((unsigned)(uintptr_t)&stage[t][tid * 4],
                          rp + (size_t)t * TILE + tid * 4);
    for (int t = 0; t < NT; ++t) {
        if (t + PIPE - 1 < NT)
            async_b128_to_lds((unsigned)(uintptr_t)&stage[(t + PIPE - 1) % PIPE][tid * 4],
                              rp + (size_t)(t + PIPE - 1) * TILE + tid * 4);
        int rem = NT - 1 - t; if (rem > PIPE - 1) rem = PIPE - 1;
        wait_async_le(rem);                     // loads complete in order: tile t done
        float4 v4 = *reinterpret_cast<const float4*>(&stage[t % PIPE][tid * 4]);
        float vv[4] = {v4.x, v4.y, v4.z, v4.w};
        #pragma unroll
        for (int c = 0; c < 4; ++c)
            atomicAdd(&hist[fkey(vv[c]) >> HSHIFT], 1u);
    }
    __syncthreads();

    // ---- threshold bin: largest bin s.t. count(keys in bins >= bin) >= TOPC ----
    unsigned psum = 0;
    #pragma unroll
    for (int i = 0; i < BPT; ++i) psum += hist[tid * BPT + i];
    part[tid] = psum;
    __syncthreads();
    if (tid == 0) {                       // exclusive suffix sum over 256 partials
        unsigned run = 0;
        for (int t = 255; t >= 0; --t) { unsigned p = part[t]; part[t] = run; run += p; }
    }
    __syncthreads();
    unsigned above = part[tid];           // count in bins >= (tid+1)*BPT
    if (above < TOPC && above + psum >= TOPC) {
        unsigned run = above;
        for (int b2 = BPT - 1; b2 >= 0; --b2) {
            run += hist[tid * BPT + b2];
            if (run >= TOPC) { sTb = tid * BPT + b2; break; }
        }
    }
    __syncthreads();
    const unsigned Tb = (unsigned)sTb;

    // ---- Pass B: collect candidates (row is L2-resident: 131 MB fits in 192 MB) ----
    const float4* r4 = reinterpret_cast<const float4*>(rp);
    for (int t = tid; t < NV / 4; t += 256) {
        if (t + 4096 < NV / 4) __builtin_prefetch(r4 + t + 4096, 0, 0);
        float4 v = r4[t];
        float a[4] = {v.x, v.y, v.z, v.w};
        #pragma unroll
        for (int c = 0; c < 4; ++c) {
            unsigned key = fkey(a[c]);
            if ((key >> HSHIFT) >= Tb) {
                int p = atomicAdd(&cnt, 1);
                if (p < CAP)
                    cand[p] = (((unsigned long long)(~key)) << 32) | (unsigned)(4 * t + c);
            }
        }
    }
    __syncthreads();
    int m = cnt; if (m > CAP) m = CAP;
    for (int i = tid; i < CAP; i += 256)
        if (i >= m) cand[i] = ~0ull;      // pads sort to the end (ascending on ~key)
    __syncthreads();

    // ---- bitonic sort ascending on packed (~key, idx) == descending by value, stable ----
    for (int k = 2; k <= CAP; k <<= 1) {
        for (int j = k >> 1; j > 0; j >>= 1) {
            for (int i = tid; i < CAP; i += 256) {
                int ixj = i ^ j;
                if (ixj > i) {
                    unsigned long long a = cand[i], b2 = cand[ixj];
                    bool up = ((i & k) == 0);
                    if (up ? (a > b2) : (a < b2)) { cand[i] = b2; cand[ixj] = a; }
                }
            }
            __syncthreads();
        }
    }

    // ---- emit top-256 (col-major so kernel 2 is coalesced) ----
    unsigned long long sv = cand[tid];
    unsigned key = ~(unsigned)(sv >> 32);
    unsigned fb  = (key & 0x80000000u) ? (key ^ 0x80000000u) : ~key;
    top_vals[tid * NB + row] = __uint_as_float(fb);
    top_idx [tid * NB + row] = (int)(unsigned)(sv & 0xFFFFFFFFu);
}

// ================= Kernel 2: per-row top-k/top-p sampling (1 block, thread b = row b) ===
__global__ __launch_bounds__(256) void sample_kernel(const float* __restrict__ params,   // [NB][3]
                                                     const float* __restrict__ top_vals,
                                                     const int*   __restrict__ top_idx,
                                                     int*         __restrict__ out) {
    __shared__ float p0s[256];
    const int b = threadIdx.x;
    int k = (int)params[b * 3 + 0]; if (k < 1) k = 1; if (k > TOPC) k = TOPC;
    const float tp   = params[b * 3 + 1];
    const float temp = params[b * 3 + 2];

    // pass 1: first softmax normalizer (top-k masked, temperature)
    const float m = top_vals[b] / temp;         // l_0 is the max (values sorted desc)
    float Z1 = 0.f;
    for (int j = 0; j < TOPC; ++j) {
        float l = ((j < k) ? top_vals[j * NB + b] : IGN) / temp;
        Z1 += expf(l - m);
    }
    const float p0 = 1.0f / Z1;                 // probs[b, 0]

    // global min over rows of probs[:,0]  (== jnp.min(cum))
    p0s[b] = p0;
    __syncthreads();
    for (int off = 128; off > 0; off >>= 1) {
        if (b < off) p0s[b] = fminf(p0s[b], p0s[b + off]);
        __syncthreads();
    }
    const float eff = fmaxf(p0s[0], tp);

    // pass 2: top-p mask (cum of first softmax), second normalizer
    unsigned keep[TOPC / 32] = {0, 0, 0, 0, 0, 0, 0, 0};
    float cum1 = 0.f, Z2 = 0.f;
    for (int j = 0; j < TOPC; ++j) {
        float l = ((j < k) ? top_vals[j * NB + b] : IGN) / temp;
        cum1 += expf(l - m) * p0;
        bool kp = (j == 0) || !(cum1 > eff);     // .at[:,0].set(False)
        if (kp) keep[j >> 5] |= 1u << (j & 31);
        Z2 += expf((kp ? l : IGN) - m);          // top_p mask writes raw -3000 (not /temp)
    }
    const float invZ2 = 1.0f / Z2;

    // pass 3: second cumsum, count positions with 0.5 > cum, pick token
    float cum2 = 0.f; int c = 0;
    for (int j = 0; j < TOPC; ++j) {
        float l  = ((j < k) ? top_vals[j * NB + b] : IGN) / temp;
        float l2 = ((keep[j >> 5] >> (j & 31)) & 1u) ? l : IGN;
        cum2 += expf(l2 - m) * invZ2;
        c += (0.5f > cum2) ? 1 : 0;
    }
    if (c > TOPC - 1) c = TOPC - 1;
    out[b] = top_idx[c * NB + b];
}

extern "C" void kernel_launch(void* const* d_in, const int* in_sizes, int n_in,
                              void* d_out, int out_size, void* d_ws, size_t ws_size,
                              hipStream_t stream) {
    (void)in_sizes; (void)n_in; (void)out_size; (void)ws_size;
    const float* logits = (const float*)d_in[0];          // [256][128000] f32
    const float* params = (const float*)d_in[1];          // [256][3] f32
    float* top_vals = (float*)d_ws;                                            // 256 KB
    int*   top_idx  = (int*)((char*)d_ws + (size_t)TOPC * NB * sizeof(float)); // 256 KB
    int*   out      = (int*)d_out;                        // 256 int32 tokens

    topk_kernel<<<NB, 256, 0, stream>>>(logits, top_vals, top_idx);
    sample_kernel<<<1, 256, 0, stream>>>(params, top_vals, top_idx, out);
}